// FeatureCorrelation_58420145160370
// MI455X (gfx1250) — compile-verified
//
#include <hip/hip_runtime.h>
#include <hip/hip_bf16.h>
#include <stdint.h>

#define EPS_PD 1e-6f
#define EPS_CS 1e-8f

#if defined(__AMDGCN__) && defined(__gfx1250__) &&                         \
    __has_builtin(__builtin_amdgcn_global_load_async_to_lds_b128) &&       \
    __has_builtin(__builtin_amdgcn_s_wait_asynccnt)
#define USE_ASYNC_LDS 1
// The builtin's params are int4 pointers in addrspace(1) (global, printed as
// __device__ by clang) and addrspace(3) (LDS / __shared__).
typedef __attribute__((ext_vector_type(4))) int v4i_t;
typedef __attribute__((address_space(1))) v4i_t gv4i_t;
typedef __attribute__((address_space(3))) v4i_t sv4i_t;
#else
#define USE_ASYNC_LDS 0
#endif

__device__ __forceinline__ float wave_reduce_sum(float v) {
#pragma unroll
  for (int off = 16; off > 0; off >>= 1) v += __shfl_xor(v, off, 32);
  return v;
}

__device__ __forceinline__ float4 comb(float4 base, float ka, float4 a,
                                       float kb, float4 b) {
  float4 o;
  o.x = fmaf(ka, a.x, fmaf(kb, b.x, base.x));
  o.y = fmaf(ka, a.y, fmaf(kb, b.y, base.y));
  o.z = fmaf(ka, a.z, fmaf(kb, b.z, base.z));
  o.w = fmaf(ka, a.w, fmaf(kb, b.w, base.w));
  return o;
}

// One 256-thread block per batch row b.  D = 2048 (harness-fixed).
__global__ __launch_bounds__(256) void FeatureCorrelation_58420145160370_kernel(
    const float* __restrict__ feat, const float* __restrict__ pos,
    float* __restrict__ out) {
  constexpr int D = 2048;
  constexpr int V4 = D / 4;  // 512 float4 per feature row
  const int tid = threadIdx.x;
  const int b = blockIdx.x;

  __shared__ float4 smem4[3 * V4];  // 24 KB staged tile (f1,f2,f3 raw)
  __shared__ float red[8][12];      // cross-wave reduction scratch

  const float4* gin = (const float4*)(feat + (size_t)b * 3 * D);
  const float4* pos4 = (const float4*)pos;

  float4 r[3][2];  // per-thread slice of f1,f2,f3 (8 floats each)

#if USE_ASYNC_LDS
  // Stage 24 KB via the CDNA5 async global->LDS DMA path (ASYNCcnt-tracked).
#pragma unroll
  for (int i = 0; i < 6; ++i) {
    const int idx = i * 256 + tid;
    __builtin_amdgcn_global_load_async_to_lds_b128(
        (gv4i_t*)(gin + idx), (sv4i_t*)(&smem4[idx]), 0, 0);
  }
  __builtin_amdgcn_s_wait_asynccnt(0);  // my wave's copies are in LDS
  __syncthreads();                      // everyone else's too
#pragma unroll
  for (int s = 0; s < 3; ++s) {
    r[s][0] = smem4[s * V4 + tid];
    r[s][1] = smem4[s * V4 + tid + 256];
  }
#else
#pragma unroll
  for (int s = 0; s < 3; ++s) {
    r[s][0] = gin[s * V4 + tid];
    r[s][1] = gin[s * V4 + tid + 256];
  }
#endif

  // x = feat + pos  (pos is 24 KB, stays L2-resident across all 8192 blocks)
#pragma unroll
  for (int s = 0; s < 3; ++s) {
#pragma unroll
    for (int j = 0; j < 2; ++j) {
      const float4 p = pos4[s * V4 + tid + j * 256];
      r[s][j].x += p.x;
      r[s][j].y += p.y;
      r[s][j].z += p.z;
      r[s][j].w += p.w;
    }
  }

  // 12 accumulators: norms, dots, and (diff^2, diff) per unordered pair.
  // sum((a-b+e)^2) = S +- 2e*M + D*e^2 recovers both signed directions.
  float n1 = 0.f, n2 = 0.f, n3 = 0.f, p12 = 0.f, p13 = 0.f, p23 = 0.f;
  float s12 = 0.f, m12 = 0.f, s13 = 0.f, m13 = 0.f, s23 = 0.f, m23 = 0.f;
#pragma unroll
  for (int j = 0; j < 2; ++j) {
    const float* a = (const float*)&r[0][j];
    const float* bb = (const float*)&r[1][j];
    const float* c = (const float*)&r[2][j];
#pragma unroll
    for (int k = 0; k < 4; ++k) {
      const float x = a[k], y = bb[k], z = c[k];
      n1 = fmaf(x, x, n1);
      n2 = fmaf(y, y, n2);
      n3 = fmaf(z, z, n3);
      p12 = fmaf(x, y, p12);
      p13 = fmaf(x, z, p13);
      p23 = fmaf(y, z, p23);
      const float t12 = x - y;
      s12 = fmaf(t12, t12, s12);
      m12 += t12;
      const float t13 = x - z;
      s13 = fmaf(t13, t13, s13);
      m13 += t13;
      const float t23 = y - z;
      s23 = fmaf(t23, t23, s23);
      m23 += t23;
    }
  }

  float acc[12] = {n1, n2, n3, p12, p13, p23, s12, m12, s13, m13, s23, m23};
#pragma unroll
  for (int k = 0; k < 12; ++k) acc[k] = wave_reduce_sum(acc[k]);

  const int lane = tid & 31, wv = tid >> 5;
  if (lane == 0) {
#pragma unroll
    for (int k = 0; k < 12; ++k) red[wv][k] = acc[k];
  }
  __syncthreads();

  float tot[12];
#pragma unroll
  for (int k = 0; k < 12; ++k) {
    float s = 0.f;
#pragma unroll
    for (int w = 0; w < 8; ++w) s += red[w][k];  // broadcast LDS reads
    tot[k] = s;
  }

  const float e2 = 2.0f * EPS_PD;
  const float corr = (float)D * EPS_PD * EPS_PD;

  const float N1 = fmaxf(sqrtf(tot[0]), EPS_CS);
  const float N2 = fmaxf(sqrtf(tot[1]), EPS_CS);
  const float N3 = fmaxf(sqrtf(tot[2]), EPS_CS);
  const float c12 = 0.5f + 0.5f * tot[3] / (N1 * N2);
  const float c13 = 0.5f + 0.5f * tot[4] / (N1 * N3);
  const float c23 = 0.5f + 0.5f * tot[5] / (N2 * N3);
  const float w12 = 1.0f / (1.0f + sqrtf(fmaxf(tot[6] + e2 * tot[7] + corr, 0.f)));
  const float w21 = 1.0f / (1.0f + sqrtf(fmaxf(tot[6] - e2 * tot[7] + corr, 0.f)));
  const float w13 = 1.0f / (1.0f + sqrtf(fmaxf(tot[8] + e2 * tot[9] + corr, 0.f)));
  const float w31 = 1.0f / (1.0f + sqrtf(fmaxf(tot[8] - e2 * tot[9] + corr, 0.f)));
  const float w23 = 1.0f / (1.0f + sqrtf(fmaxf(tot[10] + e2 * tot[11] + corr, 0.f)));
  const float w32 = 1.0f / (1.0f + sqrtf(fmaxf(tot[10] - e2 * tot[11] + corr, 0.f)));

  const float k12 = w12 + c12;  // o1 <- f2
  const float k13 = w13 + c13;  // o1 <- f3
  const float k21 = w21 + c12;  // o2 <- f1
  const float k23 = w23 + c23;  // o2 <- f3
  const float k31 = w31 + c13;  // o3 <- f1
  const float k32 = w32 + c23;  // o3 <- f2

  float4* gout = (float4*)(out + (size_t)b * 3 * D);
#pragma unroll
  for (int j = 0; j < 2; ++j) {
    const float4 A = r[0][j], Bv = r[1][j], C = r[2][j];
    const int o = tid + j * 256;
    gout[0 * V4 + o] = comb(A, k12, Bv, k13, C);
    gout[1 * V4 + o] = comb(Bv, k21, A, k23, C);
    gout[2 * V4 + o] = comb(C, k31, A, k32, Bv);
  }
}

extern "C" void kernel_launch(void* const* d_in, const int* in_sizes, int n_in,
                              void* d_out, int out_size, void* d_ws,
                              size_t ws_size, hipStream_t stream) {
  (void)n_in;
  (void)d_ws;
  (void)ws_size;
  (void)out_size;
  const float* feat = (const float*)d_in[0];
  const float* pos = (const float*)d_in[1];
  float* out = (float*)d_out;
  const int posn = in_sizes[1];      // 3*D
  const int B = in_sizes[0] / posn;  // 8192
  FeatureCorrelation_58420145160370_kernel<<<B, 256, 0, stream>>>(feat, pos, out);
}